// get_model_22136261443757
// MI455X (gfx1250) — compile-verified
//
#include <hip/hip_runtime.h>
#include <hip/hip_bf16.h>
#include <cstdint>
#include <cstddef>

#define BN_EPS 1e-5f

typedef float v2f __attribute__((ext_vector_type(2)));
typedef float v8f __attribute__((ext_vector_type(8)));

static __device__ __forceinline__ size_t gtid() {
  return (size_t)blockIdx.x * blockDim.x + threadIdx.x;
}

// ---------------- pts = transpose(xyz), l0_xyz = pts[...,:3] ----------------
__global__ void k_transpose(const float* __restrict__ xyz, float* __restrict__ pts,
                            float* __restrict__ l0, int B, int C, int N) {
  size_t t = gtid();
  size_t total = (size_t)B * N * C;
  if (t >= total) return;
  int c = (int)(t % C);
  size_t r = t / C;
  int n = (int)(r % N);
  int b = (int)(r / N);
  float v = xyz[((size_t)b * C + c) * N + n];
  pts[t] = v;                                  // (B,N,C) layout
  if (c < 3) l0[((size_t)b * N + n) * 3 + c] = v;
}

// ---------------- farthest point sampling (1 block per batch) ----------------
__global__ void k_fps(const float* __restrict__ xyz, int N, int npoint,
                      int* __restrict__ out_idx, float* __restrict__ dist) {
  __shared__ float sd[256];
  __shared__ int   si[256];
  int b = blockIdx.x;
  int tid = threadIdx.x;
  const float* X = xyz + (size_t)b * N * 3;
  float* D = dist + (size_t)b * N;
  for (int i = tid; i < N; i += blockDim.x) D[i] = 1e10f;
  __syncthreads();
  int far = 0;
  for (int s = 0; s < npoint; s++) {
    if (tid == 0) out_idx[(size_t)b * npoint + s] = far;
    float cx = X[(size_t)far * 3], cy = X[(size_t)far * 3 + 1], cz = X[(size_t)far * 3 + 2];
    float best = -1.0f;
    int bi = 0x7fffffff;
    for (int i = tid; i < N; i += blockDim.x) {
      float dx = X[(size_t)i * 3] - cx;
      float dy = X[(size_t)i * 3 + 1] - cy;
      float dz = X[(size_t)i * 3 + 2] - cz;
      float d = dx * dx + dy * dy + dz * dz;
      float dd = fminf(D[i], d);
      D[i] = dd;
      if (dd > best) { best = dd; bi = i; }
    }
    sd[tid] = best; si[tid] = bi;
    __syncthreads();
    for (int st = 128; st > 0; st >>= 1) {
      if (tid < st) {
        if (sd[tid + st] > sd[tid] ||
            (sd[tid + st] == sd[tid] && si[tid + st] < si[tid])) {
          sd[tid] = sd[tid + st];
          si[tid] = si[tid + st];
        }
      }
      __syncthreads();
    }
    far = si[0];
    __syncthreads();
  }
}

// ---------------- gather new_xyz = xyz[fps_idx] ----------------
__global__ void k_gather_xyz(const float* __restrict__ xyz, const int* __restrict__ idx,
                             int B, int N, int S, float* __restrict__ out) {
  size_t t = gtid();
  size_t total = (size_t)B * S * 3;
  if (t >= total) return;
  int c = (int)(t % 3);
  size_t r = t / 3;
  int s = (int)(r % S);
  int b = (int)(r / S);
  int id = idx[(size_t)b * S + s];
  out[t] = xyz[((size_t)b * N + id) * 3 + c];
}

// ---------------- ball query: first K indices (ascending) within radius ------
__global__ void k_ball(const float* __restrict__ xyz, const float* __restrict__ nxyz,
                       int B, int N, int S, int K, float r2, int* __restrict__ idx) {
  size_t t = gtid();
  if (t >= (size_t)B * S) return;
  int b = (int)(t / S);
  const float* X = xyz + (size_t)b * N * 3;
  float qx = nxyz[t * 3], qy = nxyz[t * 3 + 1], qz = nxyz[t * 3 + 2];
  int cnt = 0;
  int first = N - 1;
  bool have = false;
  for (int j = 0; j < N && cnt < K; j++) {
    float dx = X[(size_t)j * 3] - qx;
    float dy = X[(size_t)j * 3 + 1] - qy;
    float dz = X[(size_t)j * 3 + 2] - qz;
    if (dx * dx + dy * dy + dz * dz <= r2) {
      if (!have) { first = j; have = true; }
      idx[t * K + cnt] = j;
      cnt++;
    }
  }
  for (; cnt < K; cnt++) idx[t * K + cnt] = first;
}

// --- grouping: x = concat(xyz[idx]-new_xyz, feat[idx]) with padded stride ----
__global__ void k_group(const float* __restrict__ xyz, const float* __restrict__ nxyz,
                        const float* __restrict__ feat, const int* __restrict__ idx,
                        int B, int N, int S, int K, int Cf, int Cxp,
                        float* __restrict__ X) {
  int Cx = 3 + Cf;
  size_t t = gtid();
  size_t total = (size_t)B * S * K * Cxp;
  if (t >= total) return;
  int c = (int)(t % Cxp);
  size_t r = t / Cxp;         // ((b*S+s)*K+k)
  int k = (int)(r % K);
  size_t bs = r / K;          // b*S+s
  float v = 0.f;
  if (c < Cx) {
    int id = idx[bs * K + k];
    int b = (int)(bs / S);
    if (c < 3) v = xyz[((size_t)b * N + id) * 3 + c] - nxyz[bs * 3 + c];
    else       v = feat[((size_t)b * N + id) * Cf + (c - 3)];
  }
  X[t] = v;
}

// ---------------- pad weights: Wp[O,Cp] = W[O,C] zero-extended ----------------
__global__ void k_padw(const float* __restrict__ W, int O, int C, int Cp,
                       float* __restrict__ Wp) {
  size_t t = gtid();
  if (t >= (size_t)O * Cp) return;
  int c = (int)(t % Cp);
  int o = (int)(t / Cp);
  Wp[t] = (c < C) ? W[(size_t)o * C + c] : 0.f;
}

// ---------------- GEMM: Y[M,O] = X[M,C] * W[O,C]^T + bias ----------------
// One wave computes a 16x16 output tile via V_WMMA_F32_16X16X4_F32 (fp32 WMMA).
// Requires C % 4 == 0. Out-of-range rows/cols are clamped (their D entries are
// never stored, so clamped duplicates are harmless) -> branch-free inner loop
// with 8-byte global_load_b64 per operand fragment.
__global__ void __launch_bounds__(32)
k_gemm(const float* __restrict__ X, const float* __restrict__ W,
       const float* __restrict__ bias, float* __restrict__ Y,
       int M, int C, int O) {
  int m0 = blockIdx.x * 16, o0 = blockIdx.y * 16;
  int lane = threadIdx.x & 31;
  int half = lane >> 4;
  int l = lane & 15;
  int ra = m0 + l; ra = (ra < M) ? ra : (M - 1);
  int rb = o0 + l; rb = (rb < O) ? rb : (O - 1);
  const float* xr = X + (size_t)ra * C + half * 2;   // lanes16-31 -> K offset +2
  const float* wr = W + (size_t)rb * C + half * 2;
#if __has_builtin(__builtin_amdgcn_wmma_f32_16x16x4_f32)
  v8f acc = (v8f){0.f, 0.f, 0.f, 0.f, 0.f, 0.f, 0.f, 0.f};
  int k0 = 0;
  for (; k0 + 16 <= C; k0 += 16) {
    v2f a0 = *(const v2f*)(xr + k0);
    v2f a1 = *(const v2f*)(xr + k0 + 4);
    v2f a2 = *(const v2f*)(xr + k0 + 8);
    v2f a3 = *(const v2f*)(xr + k0 + 12);
    v2f b0 = *(const v2f*)(wr + k0);
    v2f b1 = *(const v2f*)(wr + k0 + 4);
    v2f b2 = *(const v2f*)(wr + k0 + 8);
    v2f b3 = *(const v2f*)(wr + k0 + 12);
    acc = __builtin_amdgcn_wmma_f32_16x16x4_f32(false, a0, false, b0, (short)0, acc, false, false);
    acc = __builtin_amdgcn_wmma_f32_16x16x4_f32(false, a1, false, b1, (short)0, acc, false, false);
    acc = __builtin_amdgcn_wmma_f32_16x16x4_f32(false, a2, false, b2, (short)0, acc, false, false);
    acc = __builtin_amdgcn_wmma_f32_16x16x4_f32(false, a3, false, b3, (short)0, acc, false, false);
  }
  for (; k0 < C; k0 += 4) {
    v2f a = *(const v2f*)(xr + k0);
    v2f b = *(const v2f*)(wr + k0);
    acc = __builtin_amdgcn_wmma_f32_16x16x4_f32(false, a, false, b, (short)0, acc, false, false);
  }
  int n = o0 + l;
  if (n < O) {
    float bb = bias ? bias[n] : 0.f;
    float* yc = Y + (size_t)(m0 + half * 8) * O + n;   // D: lanes0-15 M=v, lanes16-31 M=v+8
    if (m0 + 16 <= M) {
#pragma unroll
      for (int v = 0; v < 8; v++) yc[(size_t)v * O] = acc[v] + bb;
    } else {
#pragma unroll
      for (int v = 0; v < 8; v++) {
        int m = m0 + half * 8 + v;
        if (m < M) yc[(size_t)v * O] = acc[v] + bb;
      }
    }
  }
#else
  int n = o0 + l;
  for (int v = 0; v < 8; v++) {
    int m = m0 + half * 8 + v;
    if (m < M && n < O) {
      float acc = bias ? bias[n] : 0.f;
      for (int k = 0; k < C; k++) acc += X[(size_t)m * C + k] * W[(size_t)n * C + k];
      Y[(size_t)m * O + n] = acc;
    }
  }
#endif
}

// ---------------- batchnorm reductions (one block per channel) ----------------
__global__ void k_bn_mean(const float* __restrict__ Y, int M, int O, float* __restrict__ mean) {
  __shared__ float s[256];
  int o = blockIdx.x;
  float acc = 0.f;
  for (int m = threadIdx.x; m < M; m += blockDim.x) acc += Y[(size_t)m * O + o];
  s[threadIdx.x] = acc;
  __syncthreads();
  for (int st = 128; st > 0; st >>= 1) {
    if (threadIdx.x < st) s[threadIdx.x] += s[threadIdx.x + st];
    __syncthreads();
  }
  if (threadIdx.x == 0) mean[o] = s[0] / (float)M;
}

__global__ void k_bn_var(const float* __restrict__ Y, int M, int O,
                         const float* __restrict__ mean, float* __restrict__ var) {
  __shared__ float s[256];
  int o = blockIdx.x;
  float mu = mean[o];
  float acc = 0.f;
  for (int m = threadIdx.x; m < M; m += blockDim.x) {
    float d = Y[(size_t)m * O + o] - mu;
    acc += d * d;
  }
  s[threadIdx.x] = acc;
  __syncthreads();
  for (int st = 128; st > 0; st >>= 1) {
    if (threadIdx.x < st) s[threadIdx.x] += s[threadIdx.x + st];
    __syncthreads();
  }
  if (threadIdx.x == 0) var[o] = s[0] / (float)M;
}

__global__ void k_bn_apply(float* __restrict__ Y, size_t total, int O,
                           const float* __restrict__ mean, const float* __restrict__ var,
                           const float* __restrict__ g, const float* __restrict__ be) {
  size_t t = gtid();
  if (t >= total) return;
  int o = (int)(t % O);
  float x = Y[t];
  float y = g[o] * (x - mean[o]) / sqrtf(var[o] + BN_EPS) + be[o];
  Y[t] = fmaxf(y, 0.f);
}

// ---------------- max over K (axis=2) ----------------
__global__ void k_maxk(const float* __restrict__ Y, int BS, int K, int O,
                       float* __restrict__ out) {
  size_t t = gtid();
  if (t >= (size_t)BS * O) return;
  int o = (int)(t % O);
  size_t r = t / O;
  float m = Y[(size_t)(r * K) * O + o];
  for (int k = 1; k < K; k++) m = fmaxf(m, Y[(size_t)(r * K + k) * O + o]);
  out[t] = m;
}

// ---------------- FP input: concat(points1, 3-NN inverse-distance interp) ----
__global__ void k_fp_build(const float* __restrict__ xyz1, const float* __restrict__ xyz2,
                           const float* __restrict__ p1, const float* __restrict__ p2,
                           int B, int N1, int N2, int C1, int C2, float* __restrict__ X) {
  size_t t = gtid();
  if (t >= (size_t)B * N1) return;
  int b = (int)(t / N1);
  float qx = xyz1[t * 3], qy = xyz1[t * 3 + 1], qz = xyz1[t * 3 + 2];
  const float* P = xyz2 + (size_t)b * N2 * 3;
  float d0 = 3e38f, d1 = 3e38f, d2 = 3e38f;
  int i0 = 0, i1 = 0, i2 = 0;
  for (int j = 0; j < N2; j++) {
    float dx = qx - P[(size_t)j * 3];
    float dy = qy - P[(size_t)j * 3 + 1];
    float dz = qz - P[(size_t)j * 3 + 2];
    float d = dx * dx + dy * dy + dz * dz;
    if (d < d0)      { d2 = d1; i2 = i1; d1 = d0; i1 = i0; d0 = d; i0 = j; }
    else if (d < d1) { d2 = d1; i2 = i1; d1 = d;  i1 = j; }
    else if (d < d2) { d2 = d;  i2 = j; }
  }
  float w0 = 1.f / (d0 + 1e-8f);
  float w1 = 1.f / (d1 + 1e-8f);
  float w2 = 1.f / (d2 + 1e-8f);
  float ws = w0 + w1 + w2;
  w0 /= ws; w1 /= ws; w2 /= ws;
  int C = C1 + C2;
  float* xr = X + t * (size_t)C;
  for (int c = 0; c < C1; c++) xr[c] = p1[t * (size_t)C1 + c];
  const float* r0 = p2 + ((size_t)b * N2 + i0) * C2;
  const float* r1 = p2 + ((size_t)b * N2 + i1) * C2;
  const float* r2 = p2 + ((size_t)b * N2 + i2) * C2;
  for (int c = 0; c < C2; c++) xr[C1 + c] = w0 * r0[c] + w1 * r1[c] + w2 * r2[c];
}

// ---------------- log_softmax over last dim ----------------
__global__ void k_logsoftmax(const float* __restrict__ L, int rows, int C,
                             float* __restrict__ out) {
  size_t t = gtid();
  if (t >= (size_t)rows) return;
  const float* r = L + t * (size_t)C;
  float mx = r[0];
  for (int c = 1; c < C; c++) mx = fmaxf(mx, r[c]);
  float s = 0.f;
  for (int c = 0; c < C; c++) s += expf(r[c] - mx);
  float ls = logf(s);
  for (int c = 0; c < C; c++) out[t * (size_t)C + c] = r[c] - mx - ls;
}

// ---------------- l4 (B,S,C) -> out (B,C,S) ----------------
__global__ void k_l4out(const float* __restrict__ l4, int B, int S, int C,
                        float* __restrict__ out) {
  size_t t = gtid();
  size_t total = (size_t)B * S * C;
  if (t >= total) return;
  int c = (int)(t % C);
  size_t r = t / C;
  int s = (int)(r % S);
  int b = (int)(r / S);
  out[((size_t)b * C + c) * S + s] = l4[t];
}

// =======================================================================
extern "C" void kernel_launch(void* const* d_in, const int* in_sizes, int n_in,
                              void* d_out, int out_size, void* d_ws, size_t ws_size,
                              hipStream_t stream) {
  (void)in_sizes; (void)n_in; (void)out_size; (void)ws_size;
  const int B = 8, N0 = 4096, CIN = 9;
  const float* xyz = (const float*)d_in[0];
  float* out = (float*)d_out;

  // ---- bump allocator on workspace ----
  char* wsb = (char*)d_ws;
  size_t off = 0;
  auto allocf = [&](size_t n) -> float* {
    float* p = (float*)(wsb + off);
    off += ((n * sizeof(float)) + 255) & ~(size_t)255;
    return p;
  };
  auto alloci = [&](size_t n) -> int* {
    int* p = (int*)(wsb + off);
    off += ((n * sizeof(int)) + 255) & ~(size_t)255;
    return p;
  };

  float* pts   = allocf((size_t)B * N0 * CIN);
  float* l0xyz = allocf((size_t)B * N0 * 3);
  float* fdist = allocf((size_t)B * N0);
  int*   fpsix = alloci((size_t)B * 2000);
  int*   gidx  = alloci((size_t)B * 2000 * 16);
  float* l1xyz = allocf((size_t)B * 2000 * 3);
  float* l1    = allocf((size_t)B * 2000 * 32);
  float* l2xyz = allocf((size_t)B * 500 * 3);
  float* l2    = allocf((size_t)B * 500 * 64);
  float* l3xyz = allocf((size_t)B * 125 * 3);
  float* l3    = allocf((size_t)B * 125 * 128);
  float* l4xyz = allocf((size_t)B * 25 * 3);
  float* l4    = allocf((size_t)B * 25 * 256);
  float* l3p   = allocf((size_t)B * 125 * 256);
  float* l2p   = allocf((size_t)B * 500 * 256);
  float* l1p   = allocf((size_t)B * 2000 * 128);
  float* l0p   = allocf((size_t)B * 4096 * 64);
  float* meanb = allocf(256);
  float* varb  = allocf(256);
  float* wpad  = allocf(32768);                    // max 128*132 padded weights
  const size_t BIG = (size_t)B * 2000 * 16 * 32;   // 8.19M floats
  float* bufA = allocf(BIG);
  float* bufB = allocf(BIG);

  auto cdiv = [](size_t a, size_t b) { return (a + b - 1) / b; };
  auto gr1  = [&](size_t n) { return dim3((unsigned)cdiv(n, 256)); };

  // ---- GEMM + BN + ReLU: x[M,C] -> y[M,O]  (C must be multiple of 4) ----
  auto mlp_layer = [&](const float* x, float* y, size_t M, int C, int O,
                       const float* W, const float* bb, const float* g, const float* be) {
    dim3 gg((unsigned)cdiv(M, 16), (unsigned)cdiv((size_t)O, 16));
    k_gemm<<<gg, dim3(32), 0, stream>>>(x, W, bb, y, (int)M, C, O);
    k_bn_mean<<<dim3((unsigned)O), dim3(256), 0, stream>>>(y, (int)M, O, meanb);
    k_bn_var<<<dim3((unsigned)O), dim3(256), 0, stream>>>(y, (int)M, O, meanb, varb);
    k_bn_apply<<<gr1(M * (size_t)O), dim3(256), 0, stream>>>(y, M * (size_t)O, O,
                                                             meanb, varb, g, be);
  };

  // ---- set-abstraction level ----
  auto run_sa = [&](const float* in_xyz, int Nin, const float* in_feat, int Cf,
                    int S, int K, float radius, int pbase, const int* outs,
                    float* out_xyz, float* out_feat) {
    k_fps<<<dim3(B), dim3(256), 0, stream>>>(in_xyz, Nin, S, fpsix, fdist);
    k_gather_xyz<<<gr1((size_t)B * S * 3), dim3(256), 0, stream>>>(in_xyz, fpsix, B, Nin, S, out_xyz);
    k_ball<<<gr1((size_t)B * S), dim3(256), 0, stream>>>(in_xyz, out_xyz, B, Nin, S, K,
                                                         radius * radius, gidx);
    int Cx  = 3 + Cf;
    int Cxp = (Cx + 3) & ~3;                       // pad to multiple of 4, zero-filled
    size_t M = (size_t)B * S * K;
    k_group<<<gr1(M * (size_t)Cxp), dim3(256), 0, stream>>>(in_xyz, out_xyz, in_feat, gidx,
                                                            B, Nin, S, K, Cf, Cxp, bufA);
    float* x = bufA;
    float* y = bufB;
    int C = Cxp;
    for (int li = 0; li < 3; li++) {
      int O = outs[li];
      const float* W  = (const float*)d_in[pbase + li * 4 + 0];
      const float* bb = (const float*)d_in[pbase + li * 4 + 1];
      const float* g  = (const float*)d_in[pbase + li * 4 + 2];
      const float* be = (const float*)d_in[pbase + li * 4 + 3];
      if (li == 0 && Cxp != Cx) {                  // zero-pad weight rows to Cxp
        k_padw<<<gr1((size_t)O * Cxp), dim3(256), 0, stream>>>(W, O, Cx, Cxp, wpad);
        W = wpad;
      }
      mlp_layer(x, y, M, C, O, W, bb, g, be);
      float* t = x; x = y; y = t;
      C = O;
    }
    k_maxk<<<gr1((size_t)B * S * C), dim3(256), 0, stream>>>(x, B * S, K, C, out_feat);
  };

  // ---- feature propagation level (C1+C2 already multiple of 4 everywhere) ----
  auto run_fp = [&](const float* xyz1, int N1, const float* xyz2, int N2,
                    const float* p1, int C1, const float* p2, int C2,
                    int pbase, const int* outs, float* out_feat) {
    size_t M = (size_t)B * N1;
    k_fp_build<<<gr1(M), dim3(256), 0, stream>>>(xyz1, xyz2, p1, p2, B, N1, N2, C1, C2, bufA);
    float* x = bufA;
    int C = C1 + C2;
    for (int li = 0; li < 2; li++) {
      int O = outs[li];
      float* y = (li == 1) ? out_feat : bufB;
      mlp_layer(x, y, M, C, O,
                (const float*)d_in[pbase + li * 4 + 0],
                (const float*)d_in[pbase + li * 4 + 1],
                (const float*)d_in[pbase + li * 4 + 2],
                (const float*)d_in[pbase + li * 4 + 3]);
      x = y;
      C = O;
    }
  };

  // ---- pipeline ----
  k_transpose<<<gr1((size_t)B * N0 * CIN), dim3(256), 0, stream>>>(xyz, pts, l0xyz, B, CIN, N0);

  const int sa1o[3] = {16, 16, 32};
  const int sa2o[3] = {32, 32, 64};
  const int sa3o[3] = {64, 64, 128};
  const int sa4o[3] = {128, 128, 256};
  run_sa(l0xyz, N0,   pts, 9,   2000, 16, 0.1f, 1,  sa1o, l1xyz, l1);
  run_sa(l1xyz, 2000, l1,  32,  500,  8,  0.2f, 13, sa2o, l2xyz, l2);
  run_sa(l2xyz, 500,  l2,  64,  125,  8,  0.4f, 25, sa3o, l3xyz, l3);
  run_sa(l3xyz, 125,  l3,  128, 25,   8,  0.8f, 37, sa4o, l4xyz, l4);

  const int fp4o[2] = {256, 256};
  const int fp3o[2] = {256, 256};
  const int fp2o[2] = {256, 128};
  const int fp1o[2] = {128, 64};
  run_fp(l3xyz, 125,  l4xyz, 25,   l3,      128, l4,  256, 49, fp4o, l3p);
  run_fp(l2xyz, 500,  l3xyz, 125,  l2,      64,  l3p, 256, 57, fp3o, l2p);
  run_fp(l1xyz, 2000, l2xyz, 500,  l1,      32,  l2p, 256, 65, fp2o, l1p);
  run_fp(l0xyz, N0,   l1xyz, 2000, nullptr, 0,   l1p, 128, 73, fp1o, l0p);

  // ---- head: (64->32)+BN+ReLU, (32->13)+bias, log_softmax ----
  size_t Mh = (size_t)B * N0;
  mlp_layer(l0p, bufA, Mh, 64, 32,
            (const float*)d_in[81], (const float*)d_in[82],
            (const float*)d_in[83], (const float*)d_in[84]);
  {
    const float* W2 = (const float*)d_in[85];
    const float* b2 = (const float*)d_in[86];
    dim3 gg((unsigned)cdiv(Mh, 16), 1);
    k_gemm<<<gg, dim3(32), 0, stream>>>(bufA, W2, b2, bufB, (int)Mh, 32, 13);
  }
  k_logsoftmax<<<gr1(Mh), dim3(256), 0, stream>>>(bufB, (int)Mh, 13, out);

  // ---- second output: l4 transposed to (B,256,25) ----
  k_l4out<<<gr1((size_t)B * 25 * 256), dim3(256), 0, stream>>>(l4, B, 25, 256,
                                                               out + (size_t)B * N0 * 13);
}